// MACCell_68195490726280
// MI455X (gfx1250) — compile-verified
//
#include <hip/hip_runtime.h>
#include <hip/hip_bf16.h>

typedef __attribute__((ext_vector_type(2))) float v2f;
typedef __attribute__((ext_vector_type(8))) float v8f;

#define Bsz 256
#define Dsz 512
#define Lsz 32
#define HWsz 196
#define Tsz 8

static __device__ __forceinline__ v8f wmma_f32(v2f a, v2f b, v8f c) {
  // D(16x16,f32) += A(16x4,f32) x B(4x16,f32), exact fp32 matrix op (CDNA5)
  return __builtin_amdgcn_wmma_f32_16x16x4_f32(false, a, false, b, (short)0, c,
                                               false, false);
}

static __device__ __forceinline__ float wred_sum(float v) {
  #pragma unroll
  for (int o = 16; o; o >>= 1) v += __shfl_xor(v, o, 32);
  return v;
}
static __device__ __forceinline__ float wred_max(float v) {
  #pragma unroll
  for (int o = 16; o; o >>= 1) v = fmaxf(v, __shfl_xor(v, o, 32));
  return v;
}

// ---------------------------------------------------------------------------
// Batch linear: out[m,n] = sum_k [X1|X2][m,k] * W[n,k] + bias[n]
// W row-major [N, K], K = K1+K2. grid=(N/128, M/16), block=256 (8 waves).
// K loop split into two wave-uniform region loops (no per-lane selects).
// ---------------------------------------------------------------------------
__global__ __launch_bounds__(256) void k_linear(const float* __restrict__ X1,
                                                const float* __restrict__ X2,
                                                const float* __restrict__ W,
                                                const float* __restrict__ bias,
                                                float* __restrict__ out,
                                                int N, int K1, int K2) {
  const int lane = threadIdx.x & 31;
  const int wave = threadIdx.x >> 5;
  const int lrow = lane & 15;
  const int khalf = (lane >> 4) << 1;          // 0 or 2
  const int ntile = blockIdx.x * 8 + wave;
  const int mtile = blockIdx.y;
  const int K = K1 + K2;
  const int arow = mtile * 16 + lrow;
  const int ncol = ntile * 16 + lrow;
  const float* wr = W + (size_t)ncol * K;

  v8f acc = {};
  #pragma unroll 4
  for (int k0 = 0; k0 < K1; k0 += 4) {
    const int kk = k0 + khalf;
    v2f a, b;
    a.x = X1[arow * K1 + kk];
    a.y = X1[arow * K1 + kk + 1];
    b.x = wr[kk];
    b.y = wr[kk + 1];
    acc = wmma_f32(a, b, acc);
  }
  #pragma unroll 4
  for (int k0 = 0; k0 < K2; k0 += 4) {
    const int kk = k0 + khalf;
    v2f a, b;
    a.x = X2[arow * K2 + kk];
    a.y = X2[arow * K2 + kk + 1];
    b.x = wr[K1 + kk];
    b.y = wr[K1 + kk + 1];
    acc = wmma_f32(a, b, acc);
  }
  const float bn = bias[ncol];
  const int mbase = mtile * 16 + ((lane >> 4) << 3);
  #pragma unroll
  for (int r = 0; r < 8; ++r)
    out[(mbase + r) * N + ncol] = acc[r] + bn;
}

// ---------------------------------------------------------------------------
// Final linear: out = g*m + (1-g)*(Cin + msa @ Ws^T)
// ---------------------------------------------------------------------------
__global__ __launch_bounds__(256) void k_linear_final(
    const float* __restrict__ X, const float* __restrict__ W,
    const float* __restrict__ Cin, const float* __restrict__ gate,
    const float* __restrict__ Mm, float* __restrict__ out, int N, int K) {
  const int lane = threadIdx.x & 31;
  const int wave = threadIdx.x >> 5;
  const int lrow = lane & 15;
  const int khalf = (lane >> 4) << 1;
  const int ntile = blockIdx.x * 8 + wave;
  const int mtile = blockIdx.y;
  const int arow = mtile * 16 + lrow;
  const int ncol = ntile * 16 + lrow;
  const float* wr = W + (size_t)ncol * K;

  v8f acc = {};
  #pragma unroll 4
  for (int k0 = 0; k0 < K; k0 += 4) {
    const int kk = k0 + khalf;
    v2f a, b;
    a.x = X[arow * K + kk];
    a.y = X[arow * K + kk + 1];
    b.x = wr[kk];
    b.y = wr[kk + 1];
    acc = wmma_f32(a, b, acc);
  }
  const int mbase = mtile * 16 + ((lane >> 4) << 3);
  #pragma unroll
  for (int r = 0; r < 8; ++r) {
    const int m = mbase + r;
    const float g = gate[m];
    const float val = Cin[m * N + ncol] + acc[r];
    out[m * N + ncol] = g * Mm[m * N + ncol] + (1.0f - g) * val;
  }
}

// ---------------------------------------------------------------------------
// 1x1 conv as GEMM over pixels, register-tiled:
//   out[b,o,p] = sum_c W[o,c]*in[b,c,p] + bias[o], optional per-(b,o) scale.
// in layout [B, C, 196] (channel stride 196). Each wave computes a
// 64-channel x 32-pixel tile (4 M-tiles x 2 N-tiles, 8 accumulators):
// per k-step 4 A-loads + 2 B-loads feed 8 wmma (0.75 loads/wmma).
// grid=(7, B, 1), block=256: wave w covers channels [64w, 64w+64).
// ---------------------------------------------------------------------------
__global__ __launch_bounds__(256) void k_conv(const float* __restrict__ in1,
                                              const float* __restrict__ in2,
                                              const float* __restrict__ W,
                                              const float* __restrict__ bias,
                                              const float* __restrict__ scale,
                                              float* __restrict__ out,
                                              int K1, int K2) {
  const int lane = threadIdx.x & 31;
  const int wave = threadIdx.x >> 5;
  const int lrow = lane & 15;
  const int khalf = (lane >> 4) << 1;
  const int b_ = blockIdx.y;
  const int chanbase = wave * 64;
  const int K = K1 + K2;

  const int p0 = blockIdx.x * 32 + lrow;       // pixel for N-tile 0
  const int p1 = p0 + 16;                      // pixel for N-tile 1
  const int pc0 = (p0 < HWsz) ? p0 : (HWsz - 1);
  const int pc1 = (p1 < HWsz) ? p1 : (HWsz - 1);

  const float* wr0 = W + (size_t)(chanbase + lrow) * K;
  const float* wr1 = wr0 + (size_t)16 * K;
  const float* wr2 = wr0 + (size_t)32 * K;
  const float* wr3 = wr0 + (size_t)48 * K;

  v8f acc00 = {}, acc01 = {}, acc10 = {}, acc11 = {};
  v8f acc20 = {}, acc21 = {}, acc30 = {}, acc31 = {};

  // ---- region 1: channels [0, K1) from in1 ----
  {
    const float* base1 = in1 + (size_t)b_ * K1 * HWsz;
    #pragma unroll 2
    for (int k0 = 0; k0 < K1; k0 += 4) {
      const int kk = k0 + khalf;
      v2f b0, b1, a0, a1, a2, a3;
      b0.x = base1[kk * HWsz + pc0];
      b0.y = base1[(kk + 1) * HWsz + pc0];
      b1.x = base1[kk * HWsz + pc1];
      b1.y = base1[(kk + 1) * HWsz + pc1];
      a0.x = wr0[kk]; a0.y = wr0[kk + 1];
      a1.x = wr1[kk]; a1.y = wr1[kk + 1];
      a2.x = wr2[kk]; a2.y = wr2[kk + 1];
      a3.x = wr3[kk]; a3.y = wr3[kk + 1];
      acc00 = wmma_f32(a0, b0, acc00);
      acc01 = wmma_f32(a0, b1, acc01);
      acc10 = wmma_f32(a1, b0, acc10);
      acc11 = wmma_f32(a1, b1, acc11);
      acc20 = wmma_f32(a2, b0, acc20);
      acc21 = wmma_f32(a2, b1, acc21);
      acc30 = wmma_f32(a3, b0, acc30);
      acc31 = wmma_f32(a3, b1, acc31);
    }
  }
  // ---- region 2: channels [K1, K) from in2 ----
  if (K2 > 0) {
    const float* base2 = in2 + (size_t)b_ * K2 * HWsz;
    #pragma unroll 2
    for (int k0 = 0; k0 < K2; k0 += 4) {
      const int kk = k0 + khalf;
      v2f b0, b1, a0, a1, a2, a3;
      b0.x = base2[kk * HWsz + pc0];
      b0.y = base2[(kk + 1) * HWsz + pc0];
      b1.x = base2[kk * HWsz + pc1];
      b1.y = base2[(kk + 1) * HWsz + pc1];
      a0.x = wr0[K1 + kk]; a0.y = wr0[K1 + kk + 1];
      a1.x = wr1[K1 + kk]; a1.y = wr1[K1 + kk + 1];
      a2.x = wr2[K1 + kk]; a2.y = wr2[K1 + kk + 1];
      a3.x = wr3[K1 + kk]; a3.y = wr3[K1 + kk + 1];
      acc00 = wmma_f32(a0, b0, acc00);
      acc01 = wmma_f32(a0, b1, acc01);
      acc10 = wmma_f32(a1, b0, acc10);
      acc11 = wmma_f32(a1, b1, acc11);
      acc20 = wmma_f32(a2, b0, acc20);
      acc21 = wmma_f32(a2, b1, acc21);
      acc30 = wmma_f32(a3, b0, acc30);
      acc31 = wmma_f32(a3, b1, acc31);
    }
  }

  // ---- epilogue ----
  float* obuf = out + (size_t)b_ * Dsz * HWsz;
  const int rofs = (lane >> 4) << 3;
  v8f* accs[4][2] = {{&acc00, &acc01}, {&acc10, &acc11},
                     {&acc20, &acc21}, {&acc30, &acc31}};
  #pragma unroll
  for (int mt = 0; mt < 4; ++mt) {
    #pragma unroll
    for (int nt = 0; nt < 2; ++nt) {
      const int p = (nt == 0) ? p0 : p1;
      if (p < HWsz) {
        const v8f a = *accs[mt][nt];
        #pragma unroll
        for (int r = 0; r < 8; ++r) {
          const int o = chanbase + mt * 16 + rofs + r;
          float val = a[r] + bias[o];
          if (scale) val *= scale[b_ * Dsz + o];
          obuf[(size_t)o * HWsz + p] = val;
        }
      }
    }
  }
}

// ---------------------------------------------------------------------------
// ControlUnit attention: ca/softmax over L, c_new. One block (128 thr) per b.
// ---------------------------------------------------------------------------
__global__ __launch_bounds__(128) void k_control(
    const float* __restrict__ cq, const float* __restrict__ cw,
    const float* __restrict__ Wca, const float* __restrict__ bca,
    const int* __restrict__ mask, float* __restrict__ cnew_out,
    float* __restrict__ cv_out) {
  __shared__ float t[Dsz];
  __shared__ float cas[Lsz];
  __shared__ float cvs[Lsz];
  const int b_ = blockIdx.x;
  const int tid = threadIdx.x;
  const int lane = tid & 31;
  const int wave = tid >> 5;

  for (int d = tid; d < Dsz; d += 128) t[d] = cq[b_ * Dsz + d] * Wca[d];
  __syncthreads();

  for (int li = 0; li < 8; ++li) {
    const int l = wave * 8 + li;
    const float* cwp = cw + ((size_t)b_ * Lsz + l) * Dsz;
    float s = 0.0f;
    for (int d = lane; d < Dsz; d += 32) s += t[d] * cwp[d];
    s = wred_sum(s);
    if (lane == 0) {
      float v = s + bca[0];
      if (mask[b_ * Lsz + l] == 0) v = -1e30f;
      cas[l] = v;
    }
  }
  __syncthreads();

  if (tid < 32) {
    float v = cas[tid];
    const float mx = wred_max(v);
    float e = __expf(v - mx);
    const float sum = wred_sum(e);
    const float cv = e / sum;
    cvs[tid] = cv;
    cv_out[b_ * Lsz + tid] = cv;
  }
  __syncthreads();

  for (int d = tid; d < Dsz; d += 128) {
    float s = 0.0f;
    const float* cwp = cw + (size_t)b_ * Lsz * Dsz + d;
    #pragma unroll
    for (int l = 0; l < Lsz; ++l) s += cvs[l] * cwp[l * Dsz];
    cnew_out[b_ * Dsz + d] = s;
  }
}

// ---------------------------------------------------------------------------
// Channel softmax, in place: one wave per pixel, 512 channels (16 per lane).
// ---------------------------------------------------------------------------
__global__ __launch_bounds__(256) void k_softmax_ch(float* __restrict__ buf) {
  const int lane = threadIdx.x & 31;
  const int wave = threadIdx.x >> 5;
  const int g = blockIdx.x * 8 + wave;
  if (g >= Bsz * HWsz) return;
  const int b_ = g / HWsz;
  const int p = g % HWsz;
  float* base = buf + (size_t)b_ * Dsz * HWsz + p;

  float x[16];
  float mx = -1e30f;
  #pragma unroll
  for (int i = 0; i < 16; ++i) {
    x[i] = base[(lane + 32 * i) * HWsz];
    mx = fmaxf(mx, x[i]);
  }
  mx = wred_max(mx);
  float s = 0.0f;
  #pragma unroll
  for (int i = 0; i < 16; ++i) {
    x[i] = __expf(x[i] - mx);
    s += x[i];
  }
  s = wred_sum(s);
  const float inv = 1.0f / s;
  #pragma unroll
  for (int i = 0; i < 16; ++i) base[(lane + 32 * i) * HWsz] = x[i] * inv;
}

// ---------------------------------------------------------------------------
// r[b,d] = sum_p rv[b,d,p] * k[b,d,p]. One wave per (b,d).
// ---------------------------------------------------------------------------
__global__ __launch_bounds__(256) void k_rsum(const float* __restrict__ rv,
                                              const float* __restrict__ kin,
                                              float* __restrict__ r) {
  const int lane = threadIdx.x & 31;
  const int wave = threadIdx.x >> 5;
  const int id = blockIdx.x * 8 + wave;      // < 131072 exactly
  const size_t base = (size_t)id * HWsz;
  float s = 0.0f;
  for (int p = lane; p < HWsz; p += 32) s += rv[base + p] * kin[base + p];
  s = wred_sum(s);
  if (lane == 0) r[id] = s;
}

// ---------------------------------------------------------------------------
// WriteUnit attention: gate, sa over T=8, m_sa. One wave per batch row.
// ---------------------------------------------------------------------------
__global__ __launch_bounds__(32) void k_wattn(
    const float* __restrict__ cnew, const float* __restrict__ Wc,
    const float* __restrict__ bc, const float* __restrict__ cs,
    const float* __restrict__ Wsa, const float* __restrict__ bsa,
    const float* __restrict__ ms, float* __restrict__ gate,
    float* __restrict__ msa) {
  const int b_ = blockIdx.x;
  const int lane = threadIdx.x;

  float s = 0.0f;
  for (int d = lane; d < Dsz; d += 32) s += cnew[b_ * Dsz + d] * Wc[d];
  s = wred_sum(s);
  const float g = 1.0f / (1.0f + __expf(-(s + bc[0])));

  float z[Tsz];
  #pragma unroll
  for (int t = 0; t < Tsz; ++t) {
    float zz = 0.0f;
    const float* csp = cs + ((size_t)t * Bsz + b_) * Dsz;
    for (int d = lane; d < Dsz; d += 32) zz += csp[d] * Wsa[d];
    zz = wred_sum(zz);
    z[t] = g * zz + bsa[0];
  }
  float mx = -1e30f;
  #pragma unroll
  for (int t = 0; t < Tsz; ++t) mx = fmaxf(mx, z[t]);
  float sum = 0.0f;
  #pragma unroll
  for (int t = 0; t < Tsz; ++t) { z[t] = __expf(z[t] - mx); sum += z[t]; }
  const float inv = 1.0f / sum;

  #pragma unroll
  for (int i = 0; i < 16; ++i) {
    const int d = lane + 32 * i;
    float acc = 0.0f;
    #pragma unroll
    for (int t = 0; t < Tsz; ++t)
      acc += (z[t] * inv) * ms[((size_t)t * Bsz + b_) * Dsz + d];
    msa[b_ * Dsz + d] = acc;
  }
  if (lane == 0) gate[b_] = g;
}

// ---------------------------------------------------------------------------
extern "C" void kernel_launch(void* const* d_in, const int* in_sizes, int n_in,
                              void* d_out, int out_size, void* d_ws,
                              size_t ws_size, hipStream_t stream) {
  const float* c    = (const float*)d_in[0];
  const float* m    = (const float*)d_in[1];
  const float* kin  = (const float*)d_in[2];
  const float* q    = (const float*)d_in[3];
  const float* cw   = (const float*)d_in[4];
  const int*   mask = (const int*)  d_in[5];
  const float* cs   = (const float*)d_in[6];
  const float* ms   = (const float*)d_in[7];
  const float* Wcq  = (const float*)d_in[8];
  const float* bcq  = (const float*)d_in[9];
  const float* Wca  = (const float*)d_in[10];
  const float* bca  = (const float*)d_in[11];
  const float* Wm_r = (const float*)d_in[12];
  const float* bm_r = (const float*)d_in[13];
  const float* Wk   = (const float*)d_in[14];
  const float* bk   = (const float*)d_in[15];
  const float* WI   = (const float*)d_in[16];
  const float* bI   = (const float*)d_in[17];
  const float* Wra  = (const float*)d_in[18];
  const float* bra  = (const float*)d_in[19];
  const float* Wm_w = (const float*)d_in[20];
  const float* bm_w = (const float*)d_in[21];
  const float* Wsa  = (const float*)d_in[22];
  const float* bsa  = (const float*)d_in[23];
  const float* Wm2  = (const float*)d_in[24];
  const float* bm2  = (const float*)d_in[25];
  const float* Wc   = (const float*)d_in[26];
  const float* bc   = (const float*)d_in[27];
  const float* Ws   = (const float*)d_in[28];

  float* out = (float*)d_out;
  float* out_cnew = out;                     // [B,D]
  float* out_mnew = out + 131072;            // [B,D]
  float* out_cv   = out + 262144;            // [B,L]
  float* out_rv   = out + 270336;            // [B,D,H,W]

  const size_t BIG = (size_t)Bsz * Dsz * HWsz;   // 25,690,112 floats
  float* wsf    = (float*)d_ws;
  float* I_ws   = wsf;                       // conv1 out (I)
  float* u_ws   = wsf + BIG;                 // conv2 out * c_new (u)
  float* cq_ws  = wsf + 2 * BIG;
  float* mI_ws  = cq_ws + 131072;
  float* r_ws   = mI_ws + 131072;
  float* mp_ws  = r_ws + 131072;             // m_prev
  float* mu_ws  = mp_ws + 131072;            // m_ (after Wm2)
  float* msa_ws = mu_ws + 131072;
  float* g_ws   = msa_ws + 131072;           // gate [B]

  const dim3 blk256(256), blk128(128), blk32(32);
  const dim3 grid_lin(Dsz / 128, Bsz / 16);  // (4,16)
  const dim3 grid_conv(7, Bsz, 1);           // 7 pixel-tiles of 32

  // 1) cq = [c|q] @ Wcq^T + bcq
  k_linear<<<grid_lin, blk256, 0, stream>>>(c, q, Wcq, bcq, cq_ws, Dsz, Dsz, Dsz);
  // 2) control attention -> c_new (d_out), cv (d_out)
  k_control<<<Bsz, blk128, 0, stream>>>(cq_ws, cw, Wca, bca, mask, out_cnew, out_cv);
  // 3) mI = m @ Wm_r^T + bm_r
  k_linear<<<grid_lin, blk256, 0, stream>>>(m, nullptr, Wm_r, bm_r, mI_ws, Dsz, Dsz, 0);
  // 4) I = mI ⊙ (Wk·k + bk)
  k_conv<<<grid_conv, blk256, 0, stream>>>(kin, nullptr, Wk, bk, mI_ws, I_ws, Dsz, 0);
  // 5) u = (WI·[I;k] + bI) ⊙ c_new
  k_conv<<<grid_conv, blk256, 0, stream>>>(I_ws, kin, WI, bI, out_cnew, u_ws, Dsz, Dsz);
  // 6) ra = Wra·u + bra  (written into rv slot, softmaxed in place)
  k_conv<<<grid_conv, blk256, 0, stream>>>(u_ws, nullptr, Wra, bra, nullptr, out_rv, Dsz, 0);
  // 7) rv = softmax over channels (in place)
  k_softmax_ch<<<(Bsz * HWsz + 7) / 8, blk256, 0, stream>>>(out_rv);
  // 8) r[b,d] = sum_p rv*k
  k_rsum<<<Bsz * Dsz / 8, blk256, 0, stream>>>(out_rv, kin, r_ws);
  // 9) m_prev = [r|m] @ Wm_w^T + bm_w
  k_linear<<<grid_lin, blk256, 0, stream>>>(r_ws, m, Wm_w, bm_w, mp_ws, Dsz, Dsz, Dsz);
  // 10) m_ = m_prev @ Wm2^T + bm2
  k_linear<<<grid_lin, blk256, 0, stream>>>(mp_ws, nullptr, Wm2, bm2, mu_ws, Dsz, Dsz, 0);
  // 11) gate, sa over T, m_sa
  k_wattn<<<Bsz, blk32, 0, stream>>>(out_cnew, Wc, bc, cs, Wsa, bsa, ms, g_ws, msa_ws);
  // 12) m_new = g*m + (1-g)*(m_ + msa @ Ws^T)
  k_linear_final<<<grid_lin, blk256, 0, stream>>>(msa_ws, Ws, mu_ws, g_ws, m, out_mnew, Dsz, Dsz);

  (void)in_sizes; (void)n_in; (void)out_size; (void)ws_size;
}